// MAHGCNNET_32083405701286
// MI455X (gfx1250) — compile-verified
//
#include <hip/hip_runtime.h>
#include <hip/hip_bf16.h>

typedef __attribute__((ext_vector_type(2))) float v2f;
typedef __attribute__((ext_vector_type(8))) float v8f;

#define BN_EPS 1e-5f

// ---------------------------------------------------------------------------
// Stage matvec: out[b, r] = relu( w * sum_j g[b,r,j]*x[b?,j] + bias )
// wave-per-row, lanes stride columns with float4 loads (d % 4 == 0).
// x cached in LDS (<=500 floats). Bandwidth-critical kernel: streams g once.
// ---------------------------------------------------------------------------
__global__ void gcn_matvec(const float* __restrict__ g,   // [B, d, d]
                           const float* __restrict__ x,   // [*, d] (xbs = 0 or d)
                           int xbs,
                           const float* __restrict__ wptr, // nullptr => 1.0
                           const float* __restrict__ bptr, // scalar bias
                           float* __restrict__ outp,       // parts base + col offset
                           int ld,                         // 1500
                           int d) {
    __shared__ float4 sx4[128];                 // 512 floats max (d <= 500)
    const int b = blockIdx.y;
    float* sxf = (float*)sx4;
    for (int i = threadIdx.x; i < d; i += 256)
        sxf[i] = x[(size_t)b * xbs + i];
    __syncthreads();

    const int wave = threadIdx.x >> 5;
    const int lane = threadIdx.x & 31;
    const int r = blockIdx.x * 8 + wave;
    if (r >= d) return;

    const float4* grow = (const float4*)(g + ((size_t)b * d + r) * d);
    const int nv = d >> 2;
    float acc = 0.0f;
    for (int v = lane; v < nv; v += 32) {
        float4 gv = grow[v];
        float4 xv = sx4[v];
        acc += gv.x * xv.x + gv.y * xv.y + gv.z * xv.z + gv.w * xv.w;
    }
    // wave32 reduction
    for (int off = 16; off > 0; off >>= 1)
        acc += __shfl_down(acc, off, 32);
    if (lane == 0) {
        float w = wptr ? wptr[0] : 1.0f;
        float val = fmaf(w, acc, bptr[0]);
        outp[(size_t)b * ld + r] = fmaxf(val, 0.0f);
    }
}

// ---------------------------------------------------------------------------
// Atlas pooling: p[b, j] = sum_i M[i, j] * h[b, i]
// 16 batches per block (h tile in LDS), thread-per-j streaming M coalesced.
// ---------------------------------------------------------------------------
__global__ void atlas_pool(const float* __restrict__ M,    // [din, dout] row-major
                           const float* __restrict__ hsrc, // parts base + col off, ld
                           int ld,
                           float* __restrict__ p,          // [B, dout]
                           int din, int dout) {
    __shared__ float sh[16 * 512];
    const int bg = blockIdx.y;                   // batch group of 16
    for (int t = 0; t < 16; ++t)
        for (int i = threadIdx.x; i < din; i += 256)
            sh[t * din + i] = hsrc[(size_t)(bg * 16 + t) * ld + i];
    __syncthreads();

    const int j = blockIdx.x * 256 + threadIdx.x;
    if (j >= dout) return;
    float acc[16];
#pragma unroll
    for (int t = 0; t < 16; ++t) acc[t] = 0.0f;
    for (int i = 0; i < din; ++i) {
        float m = M[(size_t)i * dout + j];
#pragma unroll
        for (int t = 0; t < 16; ++t) acc[t] = fmaf(m, sh[t * din + i], acc[t]);
    }
#pragma unroll
    for (int t = 0; t < 16; ++t)
        p[(size_t)(bg * 16 + t) * dout + j] = acc[t];
}

// ---------------------------------------------------------------------------
// Training-mode BatchNorm1d over batch axis (B=256) + ReLU.
// Thread-per-feature; population variance (ddof=0) matches jnp.var.
// ---------------------------------------------------------------------------
__global__ void bn_relu(const float* __restrict__ X, const float* __restrict__ gam,
                        const float* __restrict__ bet, float* __restrict__ Y, int F) {
    const int f = blockIdx.x * 256 + threadIdx.x;
    if (f >= F) return;
    float s = 0.0f, sq = 0.0f;
    for (int b = 0; b < 256; ++b) {
        float v = X[(size_t)b * F + f];
        s += v; sq = fmaf(v, v, sq);
    }
    float m = s * (1.0f / 256.0f);
    float var = fmaxf(sq * (1.0f / 256.0f) - m * m, 0.0f);
    float inv = rsqrtf(var + BN_EPS);
    float scale = gam[f] * inv;
    float shift = bet[f] - m * scale;
    for (int b = 0; b < 256; ++b) {
        float y = fmaf(X[(size_t)b * F + f], scale, shift);
        Y[(size_t)b * F + f] = fmaxf(y, 0.0f);
    }
}

// ---------------------------------------------------------------------------
// Head GEMM: Z[m,n] = sum_k X[m,k]*W[n,k] + bias[n]   (Z = X @ W^T + b)
// One wave per 16x16 tile, V_WMMA_F32_16X16X4_F32 accumulation over K.
// A 16x4 layout: lane<16 -> {A[l][k0],A[l][k0+1]}, lane>=16 -> {+2,+3}.
// B 4x16 layout symmetric (row-striped across lanes within VGPR).
// D layout: acc[v] at (M = v + 8*(lane>>4), N = lane&15).
// Requires: M,N multiples of 16; K multiple of 4; tiles multiple of 8.
// ---------------------------------------------------------------------------
__global__ void head_gemm_wmma(const float* __restrict__ X,   // [M,K]
                               const float* __restrict__ W,   // [N,K]
                               const float* __restrict__ bias,// [N]
                               float* __restrict__ Z,         // [M,N]
                               int Mdim, int Ndim, int K) {
    const int wave = threadIdx.x >> 5;
    const int lane = threadIdx.x & 31;
    const int tilesN = Ndim >> 4;
    const int tile = blockIdx.x * 8 + wave;
    if (tile >= (Mdim >> 4) * tilesN) return;      // wave-uniform
    const int tm = (tile / tilesN) << 4;
    const int tn = (tile % tilesN) << 4;
    const int ln = lane & 15;
    const int hi = lane >> 4;

    const float* arow = X + (size_t)(tm + ln) * K + 2 * hi;
    const float* brow = W + (size_t)(tn + ln) * K + 2 * hi;

    v8f acc = {};
    for (int k0 = 0; k0 < K; k0 += 4) {
        v2f a = *(const v2f*)(arow + k0);
        v2f b = *(const v2f*)(brow + k0);
        acc = __builtin_amdgcn_wmma_f32_16x16x4_f32(
            /*neg_a=*/false, a, /*neg_b=*/false, b,
            /*c_mod=*/(short)0, acc, /*reuse_a=*/false, /*reuse_b=*/false);
    }
    const int col = tn + ln;
    const float bv = bias[col];
#pragma unroll
    for (int v = 0; v < 8; ++v)
        Z[(size_t)(tm + hi * 8 + v) * Ndim + col] = acc[v] + bv;
}

// ---------------------------------------------------------------------------
// Final layer [256,128] @ [2,128]^T + softmax -> d_out [256,2]
// ---------------------------------------------------------------------------
__global__ void head_final(const float* __restrict__ Y4, const float* __restrict__ fw4,
                           const float* __restrict__ fb4, float* __restrict__ out) {
    const int b = blockIdx.x * 256 + threadIdx.x;
    if (b >= 256) return;
    float z0 = fb4[0], z1 = fb4[1];
    for (int k = 0; k < 128; ++k) {
        float y = Y4[(size_t)b * 128 + k];
        z0 = fmaf(y, fw4[k], z0);
        z1 = fmaf(y, fw4[128 + k], z1);
    }
    float mx = fmaxf(z0, z1);
    float e0 = expf(z0 - mx), e1 = expf(z1 - mx);
    float inv = 1.0f / (e0 + e1);
    out[b * 2 + 0] = e0 * inv;
    out[b * 2 + 1] = e1 * inv;
}

// ---------------------------------------------------------------------------
extern "C" void kernel_launch(void* const* d_in, const int* in_sizes, int n_in,
                              void* d_out, int out_size, void* d_ws, size_t ws_size,
                              hipStream_t stream) {
    (void)in_sizes; (void)n_in; (void)out_size; (void)ws_size;
    const float* g1 = (const float*)d_in[0];
    const float* g2 = (const float*)d_in[1];
    const float* g3 = (const float*)d_in[2];
    const float* g4 = (const float*)d_in[3];
    const float* g5 = (const float*)d_in[4];
    const float* W1 = (const float*)d_in[5];
    const float* b1 = (const float*)d_in[6];
    const float* w2 = (const float*)d_in[7];  const float* b2 = (const float*)d_in[8];
    const float* w3 = (const float*)d_in[9];  const float* b3 = (const float*)d_in[10];
    const float* w4 = (const float*)d_in[11]; const float* b4 = (const float*)d_in[12];
    const float* w5 = (const float*)d_in[13]; const float* b5 = (const float*)d_in[14];
    const float* M1 = (const float*)d_in[15];
    const float* M2 = (const float*)d_in[16];
    const float* M3 = (const float*)d_in[17];
    const float* M4 = (const float*)d_in[18];
    const float* fw1 = (const float*)d_in[19]; const float* fb1 = (const float*)d_in[20];
    const float* fw2 = (const float*)d_in[21]; const float* fb2 = (const float*)d_in[22];
    const float* fw3 = (const float*)d_in[23]; const float* fb3 = (const float*)d_in[24];
    const float* fw4 = (const float*)d_in[25]; const float* fb4 = (const float*)d_in[26];
    const float* bn1g = (const float*)d_in[27]; const float* bn1b = (const float*)d_in[28];
    const float* bn2g = (const float*)d_in[29]; const float* bn2b = (const float*)d_in[30];
    const float* bn3g = (const float*)d_in[31]; const float* bn3b = (const float*)d_in[32];
    const float* bn4g = (const float*)d_in[33]; const float* bn4b = (const float*)d_in[34];

    // workspace carve-out (floats)
    float* ws = (float*)d_ws;
    float* X0 = ws;                 // [256,1500] parts (also h storage)
    float* Y1 = X0 + 256 * 1500;    // [256,1500]
    float* P  = Y1 + 256 * 1500;    // [256,<=400] pooled vector (reused)
    float* Z1 = P  + 256 * 400;     // [256,512]
    float* Y2 = Z1 + 256 * 512;     // [256,512]
    float* Z2 = Y2 + 256 * 512;     // [256,256]
    float* Y3 = Z2 + 256 * 256;     // [256,256]
    float* Z3 = Y3 + 256 * 256;     // [256,128]
    float* Y4 = Z3 + 256 * 128;     // [256,128]

    const dim3 blk(256);

    // Scale 500 (identity features): h1 = relu(g1 @ W1 + b1) -> parts[:, 0:500]
    gcn_matvec<<<dim3(63, 256), blk, 0, stream>>>(g1, W1, 0, nullptr, b1, X0 + 0, 1500, 500);
    atlas_pool<<<dim3(2, 16), blk, 0, stream>>>(M1, X0 + 0, 1500, P, 500, 400);
    gcn_matvec<<<dim3(50, 256), blk, 0, stream>>>(g2, P, 400, w2, b2, X0 + 500, 1500, 400);
    atlas_pool<<<dim3(2, 16), blk, 0, stream>>>(M2, X0 + 500, 1500, P, 400, 300);
    gcn_matvec<<<dim3(38, 256), blk, 0, stream>>>(g3, P, 300, w3, b3, X0 + 900, 1500, 300);
    atlas_pool<<<dim3(1, 16), blk, 0, stream>>>(M3, X0 + 900, 1500, P, 300, 200);
    gcn_matvec<<<dim3(25, 256), blk, 0, stream>>>(g4, P, 200, w4, b4, X0 + 1200, 1500, 200);
    atlas_pool<<<dim3(1, 16), blk, 0, stream>>>(M4, X0 + 1200, 1500, P, 200, 100);
    gcn_matvec<<<dim3(13, 256), blk, 0, stream>>>(g5, P, 100, w5, b5, X0 + 1400, 1500, 100);

    // MLP head with BN(+ReLU) before each linear
    bn_relu<<<dim3(6), blk, 0, stream>>>(X0, bn1g, bn1b, Y1, 1500);
    head_gemm_wmma<<<dim3(64), blk, 0, stream>>>(Y1, fw1, fb1, Z1, 256, 512, 1500);
    bn_relu<<<dim3(2), blk, 0, stream>>>(Z1, bn2g, bn2b, Y2, 512);
    head_gemm_wmma<<<dim3(32), blk, 0, stream>>>(Y2, fw2, fb2, Z2, 256, 256, 512);
    bn_relu<<<dim3(1), blk, 0, stream>>>(Z2, bn3g, bn3b, Y3, 256);
    head_gemm_wmma<<<dim3(16), blk, 0, stream>>>(Y3, fw3, fb3, Z3, 256, 128, 256);
    bn_relu<<<dim3(1), blk, 0, stream>>>(Z3, bn4g, bn4b, Y4, 128);
    head_final<<<dim3(1), blk, 0, stream>>>(Y4, fw4, fb4, (float*)d_out);
}